// DeformablePSRoIPooling_77137612636518
// MI455X (gfx1250) — compile-verified
//
#include <hip/hip_runtime.h>

// PS-ROIAlign forward for MI455X (gfx1250, wave32).
// Latency-bound gather kernel: 8x global_load_b64 per output, scalarized
// per-ROI geometry, branch-free masks, non-temporal output stores.

#define SPATIAL_SCALE 0.0625f
#define P 7          // ROI_SIZE
#define SR 2         // SAMPLING_RATIO
#define DD 32        // POOLED_DIM
#define HH 64
#define WW 64
#define CCH (DD * P * P)   // 1568 channels
#define TPB 224            // 7 wave32s; 1568 = 7 * 224

// 2-float vector with 4-byte alignment so the compiler may emit a single
// global_load_b64 at dword alignment (legal in DWORD alignment mode).
typedef float f2 __attribute__((ext_vector_type(2), aligned(4)));

__global__ __launch_bounds__(TPB)
void psroi_align_fwd(const float* __restrict__ data,
                     const float* __restrict__ rois,
                     float* __restrict__ out)
{
    const int r = blockIdx.x;                         // roi index (uniform)
    const int c = blockIdx.y * TPB + threadIdx.x;     // channel 0..1567

    // ---- per-ROI geometry: uniform across the block -> SGPR/SALU ----
    const float* __restrict__ roi = rois + (size_t)r * 5;
    const int   batch = (int)roi[0];
    const float sw = rintf(roi[1]) * SPATIAL_SCALE - 0.5f;
    const float sh = rintf(roi[2]) * SPATIAL_SCALE - 0.5f;
    const float ew = (rintf(roi[3]) + 1.0f) * SPATIAL_SCALE - 0.5f;
    const float eh = (rintf(roi[4]) + 1.0f) * SPATIAL_SCALE - 0.5f;
    const float roi_w = fmaxf(ew - sw, 0.1f);
    const float roi_h = fmaxf(eh - sh, 0.1f);
    const float bin_w = roi_w * (1.0f / P);
    const float bin_h = roi_h * (1.0f / P);
    const float sub_w = bin_w * (1.0f / SR);
    const float sub_h = bin_h * (1.0f / SR);

    // ---- per-thread position-sensitive bin decode ----
    const int d   = c / (P * P);
    const int rem = c - d * (P * P);
    const int ph  = rem / P;
    const int pw  = rem - ph * P;

    const float* __restrict__ base =
        data + ((size_t)batch * CCH + (size_t)c) * (size_t)(HH * WW);

    // ---- sample coordinates (address pass, fully unrolled) ----
    float ym[SR], ty[SR], xm[SR], tx[SR];
    int   ya[SR], xa[SR];
    #pragma unroll
    for (int s = 0; s < SR; ++s) {
        const float y = sh + (float)ph * bin_h + (float)s * sub_h;
        ym[s] = (y > -0.5f && y < (float)HH - 0.5f) ? 1.0f : 0.0f;
        const float yc = fminf(fmaxf(y, 0.0f), (float)(HH - 1));
        ya[s] = min((int)yc, HH - 2);           // yc >= 0 -> cast == floor
        ty[s] = yc - (float)ya[s];              // in [0,1]; ==1 at top clamp

        const float x = sw + (float)pw * bin_w + (float)s * sub_w;
        xm[s] = (x > -0.5f && x < (float)WW - 0.5f) ? 1.0f : 0.0f;
        const float xc = fminf(fmaxf(x, 0.0f), (float)(WW - 1));
        xa[s] = min((int)xc, WW - 2);
        tx[s] = xc - (float)xa[s];
    }

    // ---- load pass: issue all 8 b64 gathers back-to-back (max MLP) ----
    f2 row0[SR][SR], row1[SR][SR];
    #pragma unroll
    for (int iy = 0; iy < SR; ++iy) {
        #pragma unroll
        for (int ix = 0; ix < SR; ++ix) {
            const float* p = base + ya[iy] * WW + xa[ix];
            row0[iy][ix] = *(const f2*)(p);
            row1[iy][ix] = *(const f2*)(p + WW);
        }
    }

    // ---- combine pass: bilinear lerps + masked accumulation ----
    float acc = 0.0f, cnt = 0.0f;
    #pragma unroll
    for (int iy = 0; iy < SR; ++iy) {
        #pragma unroll
        for (int ix = 0; ix < SR; ++ix) {
            const f2 a = row0[iy][ix];
            const f2 b = row1[iy][ix];
            const float top = fmaf(tx[ix], a.y - a.x, a.x);
            const float bot = fmaf(tx[ix], b.y - b.x, b.x);
            const float v   = fmaf(ty[iy], bot - top, top);
            const float m   = ym[iy] * xm[ix];
            acc = fmaf(v, m, acc);
            cnt += m;
        }
    }

    const float res = acc / fmaxf(cnt, 1.0f);
    // Output is write-once (25.7 MB): non-temporal store so the 51 MB input
    // stays resident in the 192 MB L2 across all 4096 ROIs.
    __builtin_nontemporal_store(res, out + (size_t)r * CCH + (size_t)c);
}

extern "C" void kernel_launch(void* const* d_in, const int* in_sizes, int n_in,
                              void* d_out, int out_size, void* d_ws, size_t ws_size,
                              hipStream_t stream)
{
    const float* data = (const float*)d_in[0];   // (2, 1568, 64, 64) f32
    const float* rois = (const float*)d_in[1];   // (R, 5) f32
    float* out = (float*)d_out;                  // (R, 32, 7, 7) f32

    const int R = in_sizes[1] / 5;               // 4096
    dim3 grid(R, CCH / TPB, 1);                  // (4096, 7)
    dim3 block(TPB, 1, 1);                       // 224 = 7 wave32s
    psroi_align_fwd<<<grid, block, 0, stream>>>(data, rois, out);
}